// ProposedEnergyModel_44212393345445
// MI455X (gfx1250) — compile-verified
//
#include <hip/hip_runtime.h>

typedef __attribute__((ext_vector_type(2))) float v2f;
typedef __attribute__((ext_vector_type(8))) float v8f;

#define N_ATOMS 262144
#define N_MOL   8192
#define MAX_Z   100
#define DMODEL  1024

// Workspace layout (floats): ws[0..127] = ew[z] (embed @ w, padded to 128),
//                            ws[128..130] = pw = w_pos @ w
//
// Stage 1: one block, 8 waves.
//   Waves 0..6: row-tile t = wave. Compute ew[t*16 .. t*16+15] with
//   V_WMMA_F32_16X16X4_F32, accumulating over K=1024 in steps of 4.
//   A (16x4 f32): lane L holds M = L&15; VGPR0 = K base (L<16 ? 0 : 2),
//                 VGPR1 = K base + 1.
//   B (4x16 f32): column 0 only is nonzero (lane 0 holds K0,K1; lane 16 K2,K3).
//   D (16x16 f32): column 0 lives in lane 0 (M=0..7 across c[0..7]) and
//                  lane 16 (M=8..15).
//   Wave 7: lanes 0..2 compute pw[j] = dot(w_pos[j,:], w).
__global__ void precompute_kernel(const float* __restrict__ embed,
                                  const float* __restrict__ w_pos,
                                  const float* __restrict__ w,
                                  float* __restrict__ ws) {
    const int lane = threadIdx.x & 31;
    const int wave = threadIdx.x >> 5;

    if (wave < 7) {
        const int m     = lane & 15;
        const int row   = wave * 16 + m;
        const int rowC  = (row < MAX_Z) ? row : 0;         // clamp: no OOB loads
        const float rs  = (row < MAX_Z) ? 1.0f : 0.0f;     // mask padded rows
        const float bs  = (m == 0) ? 1.0f : 0.0f;          // only column N=0 of B
        const int khalf = (lane >> 4) << 1;                // 0 or 2

        v8f c = {};
        for (int k = 0; k < DMODEL; k += 4) {
            const int kb = k + khalf;
            v2f a, b;
            a.x = embed[rowC * DMODEL + kb]     * rs;
            a.y = embed[rowC * DMODEL + kb + 1] * rs;
            b.x = w[kb]     * bs;
            b.y = w[kb + 1] * bs;
            // D = A*B + C  (f32 WMMA, 16x16x4)
            c = __builtin_amdgcn_wmma_f32_16x16x4_f32(
                    false, a, false, b, (short)0, c, false, false);
        }
        if (lane == 0) {
            #pragma unroll
            for (int r = 0; r < 8; ++r) ws[wave * 16 + r] = c[r];        // M=0..7
        } else if (lane == 16) {
            #pragma unroll
            for (int r = 0; r < 8; ++r) ws[wave * 16 + 8 + r] = c[r];    // M=8..15
        }
    } else {
        if (lane < 3) {
            float acc = 0.0f;
            for (int k = 0; k < DMODEL; ++k)
                acc += w_pos[lane * DMODEL + k] * w[k];
            ws[128 + lane] = acc;
        }
    }
}

// Stage 2: out[m] = b[0]  (d_out is poisoned by the harness; must init)
__global__ void init_out_kernel(float* __restrict__ out,
                                const float* __restrict__ b) {
    const int i = blockIdx.x * blockDim.x + threadIdx.x;
    if (i < N_MOL) out[i] = b[0];
}

// Stage 3: per-atom scalar + segmented wave reduction + one atomic per run.
// N_ATOMS is a multiple of 256, so every thread is live (shuffles are safe).
__global__ void atoms_kernel(const int*   __restrict__ zn,
                             const float* __restrict__ pos,
                             const int*   __restrict__ batch,
                             const float* __restrict__ ws,
                             float*       __restrict__ out) {
    const int a    = blockIdx.x * blockDim.x + threadIdx.x;
    const int lane = threadIdx.x & 31;

    const float pw0 = ws[128], pw1 = ws[129], pw2 = ws[130];

    const int   z  = zn[a];
    const float p0 = pos[3 * a + 0];
    const float p1 = pos[3 * a + 1];
    const float p2 = pos[3 * a + 2];
    float s = ws[z] + p0 * pw0 + p1 * pw1 + p2 * pw2;

    const int seg = batch[a];   // sorted -> runs within a wave

    // Segmented inclusive scan across the wave (wave32).
    #pragma unroll
    for (int off = 1; off < 32; off <<= 1) {
        const float up    = __shfl_up(s, off, 32);
        const int   segup = __shfl_up(seg, off, 32);
        if (lane >= off && segup == seg) s += up;
    }
    // Last lane of each equal-seg run issues the atomic.
    const int segdn = __shfl_down(seg, 1, 32);
    const bool last = (lane == 31) || (segdn != seg);
    if (last) atomicAdd(&out[seg], s);
}

extern "C" void kernel_launch(void* const* d_in, const int* in_sizes, int n_in,
                              void* d_out, int out_size, void* d_ws, size_t ws_size,
                              hipStream_t stream) {
    const int*   zn    = (const int*)  d_in[0];   // atomic_numbers [262144] i32
    const float* pos   = (const float*)d_in[1];   // pos   [262144,3] f32
    const int*   batch = (const int*)  d_in[2];   // batch [262144] i32 (sorted)
    const float* embed = (const float*)d_in[3];   // embed [100,1024] f32
    const float* w_pos = (const float*)d_in[4];   // w_pos [3,1024] f32
    const float* w     = (const float*)d_in[5];   // w     [1024,1] f32
    const float* b     = (const float*)d_in[6];   // b     [1] f32

    float* out = (float*)d_out;                   // [8192,1] f32
    float* ws  = (float*)d_ws;                    // >= 131 floats

    precompute_kernel<<<1, 256, 0, stream>>>(embed, w_pos, w, ws);
    init_out_kernel<<<(N_MOL + 255) / 256, 256, 0, stream>>>(out, b);
    atoms_kernel<<<N_ATOMS / 256, 256, 0, stream>>>(zn, pos, batch, ws, out);
}